// OxDNAEnergy_32615981645830
// MI455X (gfx1250) — compile-verified
//
#include <hip/hip_runtime.h>
#include <math.h>

// ---------------- WMMA types ----------------
typedef __attribute__((ext_vector_type(2))) float v2f;
typedef __attribute__((ext_vector_type(8))) float v8f;

// ---------------- Parameter structs ----------------
struct F1C { float a, r0, shift, rlow, rhigh, blow, rclow, bhigh, rchigh; };
struct F2C { float k, r0, dc2, rlow, rhigh, blow, rclow, bhigh, rchigh; };
struct F3C { float sigma, rstar, b, rc; };
struct F4C { float a, t0, ts, b, tc; };
struct F5P { float a, xs, xc, b; };

struct Params {
  F1C f1_stck, f1_hb;
  F2C f2_crst, f2_cx;
  F3C f3_back, f3_base, f3_mix;
  F4C s4, s5, s6, h1, h2, h3, h4, h7, h8, c1, c2, c3, c4, c7, c8, x1, x4;
  F5P f5;
};

// ---------------- Host-side constant construction (double, mirrors numpy) ----------------
static inline void smoothD(double V, double dV, double r, double& b, double& rc) {
  rc = r - 2.0 * V / dV;
  b  = dV * dV / (4.0 * V);
}
static F1C mkF1(double a, double r0, double rc, double rlow, double rhigh) {
  double shift = (1.0 - exp(-a * (rc - r0))); shift *= shift;
  auto V  = [&](double r){ double t = 1.0 - exp(-a * (r - r0)); return t * t - shift; };
  auto dV = [&](double r){ double e = exp(-a * (r - r0)); return 2.0 * (1.0 - e) * a * e; };
  double blow, rclow, bhigh, rchigh;
  smoothD(V(rlow),  dV(rlow),  rlow,  blow,  rclow);
  smoothD(V(rhigh), dV(rhigh), rhigh, bhigh, rchigh);
  return F1C{(float)a,(float)r0,(float)shift,(float)rlow,(float)rhigh,
             (float)blow,(float)rclow,(float)bhigh,(float)rchigh};
}
static F2C mkF2(double k, double r0, double rc, double rlow, double rhigh) {
  double dc2 = (rc - r0) * (rc - r0);
  auto V  = [&](double r){ return 0.5 * k * ((r - r0) * (r - r0) - dc2); };
  auto dV = [&](double r){ return k * (r - r0); };
  double blow, rclow, bhigh, rchigh;
  smoothD(V(rlow),  dV(rlow),  rlow,  blow,  rclow);
  smoothD(V(rhigh), dV(rhigh), rhigh, bhigh, rchigh);
  return F2C{(float)k,(float)r0,(float)dc2,(float)rlow,(float)rhigh,
             (float)blow,(float)rclow,(float)bhigh,(float)rchigh};
}
static F3C mkF3(double sigma, double rstar) {
  double s6 = pow(sigma, 6.0);
  double V  = 4.0 * (s6 * s6 / pow(rstar, 12.0) - s6 / pow(rstar, 6.0));
  double dV = 4.0 * (-12.0 * s6 * s6 / pow(rstar, 13.0) + 6.0 * s6 / pow(rstar, 7.0));
  double b, rc;
  smoothD(V, dV, rstar, b, rc);
  return F3C{(float)sigma,(float)rstar,(float)b,(float)rc};
}
static F4C mkF4(double a, double t0, double ts) {
  double b  = a * a * ts * ts / (1.0 - a * ts * ts);
  double tc = ts + a * ts / b;
  return F4C{(float)a,(float)t0,(float)ts,(float)b,(float)tc};
}
static F5P mkF5(double a, double xs) {
  double V  = 1.0 - a * xs * xs;
  double dV = -2.0 * a * xs;
  double xc = xs - 2.0 * V / dV;
  double b  = dV * dV / (4.0 * V);
  return F5P{(float)a,(float)xs,(float)xc,(float)b};
}
static Params makeParams() {
  const double PI = 3.141592653589793;
  Params P;
  P.f1_stck = mkF1(6.0, 0.4, 0.9,  0.32, 0.75);
  P.f1_hb   = mkF1(8.0, 0.4, 0.75, 0.34, 0.7);
  P.f2_crst = mkF2(47.5, 0.575, 0.675, 0.495, 0.655);
  P.f2_cx   = mkF2(46.0, 0.4,   0.6,   0.22,  0.58);
  P.f3_back = mkF3(0.7,   0.675);
  P.f3_base = mkF3(0.33,  0.32);
  P.f3_mix  = mkF3(0.515, 0.5);
  P.s4 = mkF4(1.3, 0.0, 0.8);
  P.s5 = mkF4(0.9, 0.0, 0.95);
  P.s6 = mkF4(0.9, 0.0, 0.95);
  P.h1 = mkF4(1.5, 0.0, 0.7);
  P.h2 = mkF4(1.5, 0.0, 0.7);
  P.h3 = mkF4(1.5, 0.0, 0.7);
  P.h4 = mkF4(0.46, PI, 0.7);
  P.h7 = mkF4(4.0, 0.5 * PI, 0.45);
  P.h8 = mkF4(4.0, 0.5 * PI, 0.45);
  P.c1 = mkF4(2.25, 0.791, 0.58);
  P.c2 = mkF4(1.7, 1.003, 0.68);
  P.c3 = mkF4(1.7, 1.003, 0.68);
  P.c4 = mkF4(1.5, 0.0, 0.7);
  P.c7 = mkF4(1.7, 0.555, 0.68);
  P.c8 = mkF4(1.7, 0.555, 0.68);
  P.x1 = mkF4(2.0, 2.592, 0.65);
  P.x4 = mkF4(1.3, 0.0, 0.8);
  P.f5 = mkF5(2.0, -0.65);
  return P;
}

// ---------------- Device vector helpers ----------------
struct V3 { float x, y, z; };
__device__ __forceinline__ V3 vadd(V3 a, V3 b){ return {a.x+b.x, a.y+b.y, a.z+b.z}; }
__device__ __forceinline__ V3 vsub(V3 a, V3 b){ return {a.x-b.x, a.y-b.y, a.z-b.z}; }
__device__ __forceinline__ V3 vscale(float s, V3 a){ return {s*a.x, s*a.y, s*a.z}; }
__device__ __forceinline__ float dot3(V3 a, V3 b){ return a.x*b.x + a.y*b.y + a.z*b.z; }
__device__ __forceinline__ V3 minimg(V3 d, V3 b){
  return { d.x - b.x*rintf(d.x/b.x), d.y - b.y*rintf(d.y/b.y), d.z - b.z*rintf(d.z/b.z) };
}
__device__ __forceinline__ float vnorm(V3 d){ return sqrtf(dot3(d,d) + 1e-12f); }
__device__ __forceinline__ float acosc(float c){
  return acosf(fminf(fmaxf(c, -1.0f + 1e-6f), 1.0f - 1e-6f));
}
__device__ __forceinline__ void quat_axes(float4 q, V3& a1, V3& a2, V3& a3){
  float w=q.x, x=q.y, y=q.z, z=q.w;
  a1 = { w*w + x*x - y*y - z*z, 2.0f*(x*y + w*z),        2.0f*(x*z - w*y) };
  a2 = { 2.0f*(x*y - w*z),      w*w - x*x + y*y - z*z,   2.0f*(y*z + w*x) };
  a3 = { 2.0f*(x*z + w*y),      2.0f*(y*z - w*x),        w*w - x*x - y*y + z*z };
}

// ---------------- Potential pieces ----------------
__device__ __forceinline__ float f1(float r, const F1C& c){
  float e  = expf(-c.a * (r - c.r0));
  float om = 1.0f - e;
  float core = om*om - c.shift;
  float dl = r - c.rclow, dh = r - c.rchigh;
  float v = (r < c.rlow) ? c.blow*dl*dl : ((r < c.rhigh) ? core : c.bhigh*dh*dh);
  return (r > c.rclow && r < c.rchigh) ? v : 0.0f;
}
__device__ __forceinline__ float f2(float r, const F2C& c){
  float d0 = r - c.r0;
  float core = 0.5f * c.k * (d0*d0 - c.dc2);
  float dl = r - c.rclow, dh = r - c.rchigh;
  float v = (r < c.rlow) ? c.blow*dl*dl : ((r < c.rhigh) ? core : c.bhigh*dh*dh);
  return (r > c.rclow && r < c.rchigh) ? v : 0.0f;
}
__device__ __forceinline__ float f3(float r, const F3C& c, float eps){
  float x  = c.sigma / r;
  float x2 = x*x;
  float x6 = x2*x2*x2;
  float lj = 4.0f * (x6*x6 - x6);
  float d  = r - c.rc;
  float v  = (r < c.rstar) ? lj : c.b*d*d;
  return eps * ((r < c.rc) ? v : 0.0f);
}
__device__ __forceinline__ float f4(float th, const F4C& c){
  float dt = fabsf(th - c.t0);
  float dc = c.tc - dt;
  float v = (dt < c.ts) ? (1.0f - c.a*dt*dt) : (c.b*dc*dc);
  return (dt < c.tc) ? v : 0.0f;
}
__device__ __forceinline__ float f5(float x, const F5P& c){
  float dc = c.xc - x;
  float v = (x > c.xs) ? (1.0f - c.a*x*x) : (c.b*dc*dc);
  v = (x > 0.0f) ? 1.0f : v;
  return (x > c.xc) ? v : 0.0f;
}

// ---------------- WMMA-based deterministic reductions ----------------
// Wave reduce with V_WMMA_F32_16X16X4_F32: A[m][0]=partial(lane m), A[m][2]=partial(lane m+16),
// B = ones  =>  D[m][n] = partial[m] + partial[m+16] (replicated over n).
// Sum of a lane's 8 D regs + shfl_xor(16) gives the full 32-lane sum in every lane. Exact f32.
__device__ __forceinline__ float wave_reduce_wmma(float acc){
  v2f a; a[0] = acc; a[1] = 0.0f;
  v2f b; b[0] = 1.0f; b[1] = 1.0f;
  v8f c = {};
  c = __builtin_amdgcn_wmma_f32_16x16x4_f32(false, a, false, b, (short)0, c, false, false);
  float s = ((c[0] + c[1]) + (c[2] + c[3])) + ((c[4] + c[5]) + (c[6] + c[7]));
  s += __shfl_xor(s, 16, 32);
  return s;
}

// Block partial -> fixed-order sum by thread 0 (deterministic). blockDim.x must be 256.
__device__ __forceinline__ float block_reduce_wmma(float acc){
  __shared__ float wsum[8];
  float s = wave_reduce_wmma(acc);
  int lane = threadIdx.x & 31;
  int wid  = threadIdx.x >> 5;
  if (lane == 0) wsum[wid] = s;
  __syncthreads();
  float tot = 0.0f;
  if (threadIdx.x == 0) {
    int nw = blockDim.x >> 5;
    for (int k = 0; k < nw; ++k) tot += wsum[k];
  }
  return tot; // valid in thread 0
}

// ---------------- Kernels ----------------
__global__ void oxdna_bonded_kernel(const float* __restrict__ pos,
                                    const float4* __restrict__ quat,
                                    const float* __restrict__ boxp,
                                    const float* __restrict__ seps,
                                    const float* __restrict__ exclp,
                                    const int2* __restrict__ pairs,
                                    int np, float* __restrict__ ws, Params P){
  const float excl = *exclp;
  const V3 box = { boxp[0], boxp[1], boxp[2] };
  float acc = 0.0f;
  for (int p = blockIdx.x * blockDim.x + threadIdx.x; p < np; p += gridDim.x * blockDim.x) {
    int2 ij = pairs[p];
    int i = ij.x, j = ij.y;
    V3 pi = { pos[3*i], pos[3*i+1], pos[3*i+2] };
    V3 pj = { pos[3*j], pos[3*j+1], pos[3*j+2] };
    V3 a1i, a2i, a3i, a1j, a2j, a3j;
    quat_axes(quat[i], a1i, a2i, a3i);
    quat_axes(quat[j], a1j, a2j, a3j);
    V3 backi = vscale(-0.4f, a1i), backj = vscale(-0.4f, a1j);
    V3 stcki = vscale( 0.34f, a1i), stckj = vscale( 0.34f, a1j);
    V3 basei = vscale( 0.4f, a1i), basej = vscale( 0.4f, a1j);

    // disp(si,sj) = min_image(pj + sj - (pi + si))
    V3 db = minimg(vsub(vadd(pj, backj), vadd(pi, backi)), box);
    float rb = vnorm(db);
    // FENE
    float t = (rb - 0.7525f) * (1.0f / 0.25f);
    float arg = fminf(t * t, 1.0f - 1e-6f);
    arg = fmaxf(arg, 0.0f);
    acc += -log1pf(-arg); // -0.5 * 2.0 * log1p(-arg)

    // bonded excluded volume
    acc += f3(vnorm(minimg(vsub(vadd(pj, basej), vadd(pi, basei)), box)), P.f3_base, excl);
    acc += f3(vnorm(minimg(vsub(vadd(pj, basej), vadd(pi, backi)), box)), P.f3_mix,  excl);
    acc += f3(vnorm(minimg(vsub(vadd(pj, backj), vadd(pi, basei)), box)), P.f3_mix,  excl);

    // stacking
    V3 ds = minimg(vsub(vadd(pj, stckj), vadd(pi, stcki)), box);
    float rs = vnorm(ds);
    float inv_rs = 1.0f / rs;
    V3 rhs = vscale(inv_rs, ds);
    float inv_rb = 1.0f / rb;
    V3 rhb = vscale(inv_rb, db);
    float th4 = acosc(dot3(a3i, a3j));
    float th5 = acosc(dot3(rhs, a3j));
    float th6 = acosc(-dot3(rhs, a3i));
    float es = seps[p] * f1(rs, P.f1_stck)
             * f4(th4, P.s4) * f4(th5, P.s5) * f4(th6, P.s6)
             * f5(dot3(a2i, rhb), P.f5) * f5(dot3(a2j, rhb), P.f5);
    acc += es;
  }
  float tot = block_reduce_wmma(acc);
  if (threadIdx.x == 0) ws[blockIdx.x] = tot;
}

__global__ void oxdna_nonbonded_kernel(const float* __restrict__ pos,
                                       const float4* __restrict__ quat,
                                       const float* __restrict__ boxp,
                                       const float* __restrict__ hbm,
                                       const float* __restrict__ exclp,
                                       const int2* __restrict__ pairs,
                                       const int* __restrict__ btypes,
                                       int np, float* __restrict__ ws, Params P){
  const float excl = *exclp;
  const V3 box = { boxp[0], boxp[1], boxp[2] };
  float acc = 0.0f;
  for (int p = blockIdx.x * blockDim.x + threadIdx.x; p < np; p += gridDim.x * blockDim.x) {
    int2 ij = pairs[p];
    int i = ij.x, j = ij.y;
    V3 pi = { pos[3*i], pos[3*i+1], pos[3*i+2] };
    V3 pj = { pos[3*j], pos[3*j+1], pos[3*j+2] };
    V3 a1i, a2i, a3i, a1j, a2j, a3j;
    quat_axes(quat[i], a1i, a2i, a3i);
    quat_axes(quat[j], a1j, a2j, a3j);
    V3 backi = vscale(-0.4f, a1i), backj = vscale(-0.4f, a1j);
    V3 stcki = vscale( 0.34f, a1i), stckj = vscale( 0.34f, a1j);
    V3 basei = vscale( 0.4f, a1i), basej = vscale( 0.4f, a1j);

    // nonbonded excluded volume
    acc += f3(vnorm(minimg(vsub(vadd(pj, backj), vadd(pi, backi)), box)), P.f3_back, excl);
    acc += f3(vnorm(minimg(vsub(vadd(pj, basej), vadd(pi, backi)), box)), P.f3_mix,  excl);
    acc += f3(vnorm(minimg(vsub(vadd(pj, backj), vadd(pi, basei)), box)), P.f3_mix,  excl);

    // base-base displacement (shared by exc, HB, cross-stack)
    V3 dh = minimg(vsub(vadd(pj, basej), vadd(pi, basei)), box);
    float rh = vnorm(dh);
    acc += f3(rh, P.f3_base, excl);
    float inv_rh = 1.0f / rh;
    V3 rhh = vscale(inv_rh, dh);

    float t1 = acosc(-dot3(a1i, a1j));
    float t2 = acosc(-dot3(rhh, a1j));
    float t3 = acosc( dot3(rhh, a1i));
    float t4 = acosc( dot3(a3i, a3j));
    float t7 = acosc(-dot3(rhh, a3j));
    float t8 = acosc( dot3(rhh, a3i));

    // hydrogen bonding
    float eps_hb = hbm[btypes[i] * 4 + btypes[j]];
    acc += eps_hb * f1(rh, P.f1_hb)
         * f4(t1, P.h1) * f4(t2, P.h2) * f4(t3, P.h3)
         * f4(t4, P.h4) * f4(t7, P.h7) * f4(t8, P.h8);

    // cross-stacking
    acc += f2(rh, P.f2_crst)
         * f4(t1, P.c1) * f4(t2, P.c2) * f4(t3, P.c3)
         * f4(t4, P.c4) * f4(t7, P.c7) * f4(t8, P.c8);

    // coaxial stacking
    V3 dx = minimg(vsub(vadd(pj, stckj), vadd(pi, stcki)), box);
    float rx = vnorm(dx);
    float inv_rx = 1.0f / rx;
    V3 rhx = vscale(inv_rx, dx);
    acc += f2(rx, P.f2_cx) * f4(t1, P.x1) * f4(t4, P.x4)
         * f5(dot3(a2i, rhx), P.f5) * f5(dot3(a2j, rhx), P.f5);
  }
  float tot = block_reduce_wmma(acc);
  if (threadIdx.x == 0) ws[blockIdx.x] = tot;
}

__global__ void oxdna_final_reduce_kernel(const float* __restrict__ ws, int n,
                                          float* __restrict__ out){
  float acc = 0.0f;
  for (int i = threadIdx.x; i < n; i += blockDim.x) acc += ws[i];
  float tot = block_reduce_wmma(acc);
  if (threadIdx.x == 0) out[0] = tot;
}

// ---------------- Launch ----------------
extern "C" void kernel_launch(void* const* d_in, const int* in_sizes, int n_in,
                              void* d_out, int out_size, void* d_ws, size_t ws_size,
                              hipStream_t stream) {
  (void)n_in; (void)out_size; (void)ws_size;
  const float*  pos  = (const float*)d_in[0];
  const float4* quat = (const float4*)d_in[1];
  const float*  box  = (const float*)d_in[2];
  const float*  seps = (const float*)d_in[3];
  const float*  hbm  = (const float*)d_in[4];
  const float*  excl = (const float*)d_in[5];
  const int2*   bp   = (const int2*)d_in[6];
  const int2*   nbp  = (const int2*)d_in[7];
  const int*    bt   = (const int*)d_in[8];

  const int NB = in_sizes[6] / 2;  // bonded pair count
  const int NN = in_sizes[7] / 2;  // nonbonded pair count

  Params P = makeParams();
  float* ws = (float*)d_ws;

  const int T  = 256;   // 8 waves of 32
  const int B1 = 256;   // bonded blocks
  const int B2 = 1024;  // nonbonded blocks (~4 pairs/thread)

  oxdna_bonded_kernel<<<B1, T, 0, stream>>>(pos, quat, box, seps, excl, bp, NB, ws, P);
  oxdna_nonbonded_kernel<<<B2, T, 0, stream>>>(pos, quat, box, hbm, excl, nbp, bt, NN, ws + B1, P);
  oxdna_final_reduce_kernel<<<1, T, 0, stream>>>(ws, B1 + B2, (float*)d_out);
}